// PointTransformerLayer_12996571038153
// MI455X (gfx1250) — compile-verified
//
#include <hip/hip_runtime.h>
#include <hip/hip_bf16.h>

// ---------------------------------------------------------------------------
// Point Transformer layer, fused for MI455X (gfx1250, wave32, WMMA).
// B=8, N=4096, d_points=64, d_model=128, K=16 neighbors.
// ---------------------------------------------------------------------------

typedef float v2f __attribute__((ext_vector_type(2)));
typedef float v8f __attribute__((ext_vector_type(8)));

#define NPTS   4096
#define BATCH  8
#define DMODEL 128
#define KNN    16
#define ROWS   ((size_t)BATCH * NPTS)          // 32768
#define RS     132     // padded LDS row stride for attention A-staging (floats)
#define KC     32      // K-chunk staged in LDS per GEMM iteration
#define WPAD   144     // padded LDS row stride for staged weights (floats)
                       // bank = (16*k + c) % 64 -> rows k and k+2 disjoint

__device__ __forceinline__ v8f wmma4(v2f a, v2f b, v8f c) {
  // D = A(16x4 f32) * B(4x16 f32) + C(16x16 f32)
  return __builtin_amdgcn_wmma_f32_16x16x4_f32(
      /*neg_a=*/false, a, /*neg_b=*/false, b,
      /*c_mod=*/(short)0, c, /*reuse_a=*/false, /*reuse_b=*/false);
}

// ---------------------------------------------------------------------------
// KNN: per (batch, point) top-16 smallest squared distances (includes self).
// 256 threads/block, one query per thread, candidates tiled through LDS.
// ---------------------------------------------------------------------------
__global__ void knn_kernel(const float* __restrict__ xyz, int* __restrict__ knn) {
  __shared__ float sx[256], sy[256], sz[256];
  const int b = blockIdx.y;
  const int n = blockIdx.x * 256 + threadIdx.x;
  const float* base = xyz + (size_t)b * NPTS * 3;
  const float qx = base[n * 3 + 0];
  const float qy = base[n * 3 + 1];
  const float qz = base[n * 3 + 2];

  float dist[KNN];
  int   idxv[KNN];
#pragma unroll
  for (int i = 0; i < KNN; ++i) { dist[i] = 3.0e38f; idxv[i] = 0; }

  for (int c0 = 0; c0 < NPTS; c0 += 256) {
    __syncthreads();
    const int ci = c0 + threadIdx.x;
    sx[threadIdx.x] = base[ci * 3 + 0];
    sy[threadIdx.x] = base[ci * 3 + 1];
    sz[threadIdx.x] = base[ci * 3 + 2];
    __syncthreads();
#pragma unroll 4
    for (int j = 0; j < 256; ++j) {
      const float dx = qx - sx[j];
      const float dy = qy - sy[j];
      const float dz = qz - sz[j];
      const float d  = fmaf(dx, dx, fmaf(dy, dy, dz * dz));
      if (d < dist[KNN - 1]) {                  // strict < ~ stable argsort
        dist[KNN - 1] = d; idxv[KNN - 1] = c0 + j;
#pragma unroll
        for (int t = KNN - 1; t > 0; --t) {
          if (dist[t] < dist[t - 1]) {
            float td = dist[t]; dist[t] = dist[t - 1]; dist[t - 1] = td;
            int   ti = idxv[t]; idxv[t] = idxv[t - 1]; idxv[t - 1] = ti;
          }
        }
      }
    }
  }
  int* o = knn + ((size_t)b * NPTS + n) * KNN;
#pragma unroll
  for (int i = 0; i < KNN; ++i) o[i] = idxv[i];
}

// ---------------------------------------------------------------------------
// Generic row-major WMMA GEMM: out(M x 128) = A(M x K) @ W(K x 128) [+ bias].
// 8 waves/block, one 16-row tile per wave, all 128 columns (8 accum tiles).
// Weights are staged through LDS in 32-row chunks (conflict-free pad stride)
// so B fragments come from ds_load instead of per-wave global traffic.
// ---------------------------------------------------------------------------
__global__ void wmma_gemm_rm(const float* __restrict__ A,
                             const float* __restrict__ W,
                             const float* __restrict__ bias,
                             float* __restrict__ out, int M, int K) {
  __shared__ float sW[KC * WPAD];               // 18 KB

  const int tid  = threadIdx.x;
  const int wave = (int)((blockIdx.x * blockDim.x + tid) >> 5);
  const int lane = tid & 31;
  const int m0   = wave * 16;
  const bool valid = (m0 < M);
  const int half = lane >> 4;        // 0: K offsets {0,1}; 1: {2,3}
  const int ln   = lane & 15;        // A row / C column-within-tile

  v8f acc[8];
#pragma unroll
  for (int t = 0; t < 8; ++t) {
    const float bv = bias ? bias[t * 16 + ln] : 0.0f;
#pragma unroll
    for (int r = 0; r < 8; ++r) acc[t][r] = bv;
  }

  const float* Arow = valid ? (A + (size_t)(m0 + ln) * K) : A;

  // staging map: thread covers row (tid>>3), 16 cols starting at (tid&7)*16
  const int srow = tid >> 3;
  const int scol = (tid & 7) * 16;

  for (int kc = 0; kc < K; kc += KC) {
    __syncthreads();
    const float4* gsrc = (const float4*)(W + (size_t)(kc + srow) * DMODEL + scol);
    float4* ldst = (float4*)(sW + srow * WPAD + scol);
#pragma unroll
    for (int i = 0; i < 4; ++i) ldst[i] = gsrc[i];
    if (kc + KC < K)  // hint next chunk toward the caches
      __builtin_prefetch(W + (size_t)(kc + KC + srow) * DMODEL + scol, 0, 1);
    __syncthreads();

    if (valid) {
#pragma unroll
      for (int k0 = 0; k0 < KC; k0 += 4) {
        const int ka = k0 + half * 2;
        v2f a;
        a.x = Arow[kc + ka];
        a.y = Arow[kc + ka + 1];
#pragma unroll
        for (int t = 0; t < 8; ++t) {
          const int c = t * 16 + ln;
          v2f bf;
          bf.x = sW[ka * WPAD + c];
          bf.y = sW[(ka + 1) * WPAD + c];
          acc[t] = wmma4(a, bf, acc[t]);
        }
      }
    }
  }

  if (valid) {
#pragma unroll
    for (int t = 0; t < 8; ++t)
#pragma unroll
      for (int r = 0; r < 8; ++r)
        out[(size_t)(m0 + r + 8 * half) * DMODEL + t * 16 + ln] = acc[t][r];
  }
}

// ---------------------------------------------------------------------------
// Shared WMMA helper for the attention kernel: acc += LDS_A(16x128) @ Wg(128x128)
// ---------------------------------------------------------------------------
__device__ __forceinline__ void wmma_tile_acc(const float* __restrict__ Wg,
                                              const float* hA, int ln, int half,
                                              v8f acc[8]) {
#pragma unroll 4
  for (int k0 = 0; k0 < DMODEL; k0 += 4) {
    const int ka = k0 + half * 2;
    v2f a;
    a.x = hA[ln * RS + ka];
    a.y = hA[ln * RS + ka + 1];
#pragma unroll
    for (int t = 0; t < 8; ++t) {
      const int c = t * 16 + ln;
      v2f bf;
      bf.x = Wg[(size_t)ka * DMODEL + c];
      bf.y = Wg[(size_t)(ka + 1) * DMODEL + c];
      acc[t] = wmma4(a, bf, acc[t]);
    }
  }
}

// ---------------------------------------------------------------------------
// Fused attention: one wave per point. pos_enc -> h -> relu(h@g1+b)@g2+b ->
// softmax over 16 neighbors -> agg = sum attn*(vf+pos_enc). Output agg (M x 128).
// ---------------------------------------------------------------------------
__global__ void pt_attn_kernel(const float* __restrict__ xyz,
                               const int*   __restrict__ knn,
                               const float* __restrict__ qf,
                               const float* __restrict__ kfeat,
                               const float* __restrict__ vfeat,
                               const float* __restrict__ d1w, const float* __restrict__ d1b,
                               const float* __restrict__ d2w, const float* __restrict__ d2b,
                               const float* __restrict__ g1w, const float* __restrict__ g1b,
                               const float* __restrict__ g2w, const float* __restrict__ g2b,
                               float* __restrict__ agg) {
  __shared__ float shA[4][16 * RS];   // per-wave 16x128 A-staging (padded)
  __shared__ float srel[4][16][3];
  __shared__ int   snb[4][16];

  const int w    = threadIdx.x >> 5;
  const int lane = threadIdx.x & 31;
  const int p    = blockIdx.x * 4 + w;          // 0 .. 32767
  const int b    = p >> 12;
  const int n    = p & (NPTS - 1);
  const int half = lane >> 4;
  const int ln   = lane & 15;
  float* hA = shA[w];

  // --- neighbor indices + relative positions (lanes 0..15, one neighbor each)
  if (lane < KNN) {
    const int nb = knn[(size_t)p * KNN + lane];
    snb[w][lane] = nb;
    const float* xb = xyz + (size_t)b * NPTS * 3;
    srel[w][lane][0] = xb[n * 3 + 0] - xb[nb * 3 + 0];
    srel[w][lane][1] = xb[n * 3 + 1] - xb[nb * 3 + 1];
    srel[w][lane][2] = xb[n * 3 + 2] - xb[nb * 3 + 2];
  }
  __syncthreads();

  // --- t1 = relu(rel @ d1_w + d1_b)  -> LDS (16 x 128)
#pragma unroll
  for (int cc = 0; cc < 4; ++cc) {
    const int c = lane + cc * 32;
    const float w0 = d1w[c], w1 = d1w[DMODEL + c], w2 = d1w[2 * DMODEL + c];
    const float bb = d1b[c];
#pragma unroll
    for (int m = 0; m < 16; ++m) {
      float v = fmaf(srel[w][m][0], w0,
                fmaf(srel[w][m][1], w1,
                fmaf(srel[w][m][2], w2, bb)));
      hA[m * RS + c] = fmaxf(v, 0.0f);
    }
  }
  __syncthreads();

  // --- pos_enc = t1 @ d2_w + d2_b   (C-layout regs)
  v8f pos[8];
#pragma unroll
  for (int t = 0; t < 8; ++t) {
    const float bv = d2b[t * 16 + ln];
#pragma unroll
    for (int r = 0; r < 8; ++r) pos[t][r] = bv;
  }
  wmma_tile_acc(d2w, hA, ln, half, pos);
  __syncthreads();

  // --- gather row bases for this lane's 8 neighbor rows (j = r + 8*half)
  size_t rb[8];
#pragma unroll
  for (int r = 0; r < 8; ++r)
    rb[r] = ((size_t)b * NPTS + snb[w][r + 8 * half]) * DMODEL;

  // --- h = q - kf + pos_enc  -> LDS (A-layout staging)
#pragma unroll
  for (int t = 0; t < 8; ++t) {
    const int c = t * 16 + ln;
    const float qv = qf[(size_t)p * DMODEL + c];
#pragma unroll
    for (int r = 0; r < 8; ++r) {
      const float hx = qv - kfeat[rb[r] + c] + pos[t][r];
      hA[(r + 8 * half) * RS + c] = hx;
    }
  }
  __syncthreads();

  // --- t2 = relu(h @ g1 + g1_b)
  v8f t2[8];
#pragma unroll
  for (int t = 0; t < 8; ++t) {
    const float bv = g1b[t * 16 + ln];
#pragma unroll
    for (int r = 0; r < 8; ++r) t2[t][r] = bv;
  }
  wmma_tile_acc(g1w, hA, ln, half, t2);
  __syncthreads();
#pragma unroll
  for (int t = 0; t < 8; ++t) {
    const int c = t * 16 + ln;
#pragma unroll
    for (int r = 0; r < 8; ++r)
      hA[(r + 8 * half) * RS + c] = fmaxf(t2[t][r], 0.0f);
  }
  __syncthreads();

  // --- attn_logits = t2 @ g2 + g2_b
  v8f at[8];
#pragma unroll
  for (int t = 0; t < 8; ++t) {
    const float bv = g2b[t * 16 + ln];
#pragma unroll
    for (int r = 0; r < 8; ++r) at[t][r] = bv;
  }
  wmma_tile_acc(g2w, hA, ln, half, at);

  // --- softmax over the 16 neighbors per channel (rows of the C tile).
  // Lane c holds rows 0..7 of column c, lane c+16 holds rows 8..15.
  const float scale = 0.0883883476483184f;   // 1/sqrt(128)
#pragma unroll
  for (int t = 0; t < 8; ++t) {
    float mx = -3.0e38f;
#pragma unroll
    for (int r = 0; r < 8; ++r) {
      at[t][r] *= scale;
      mx = fmaxf(mx, at[t][r]);
    }
    mx = fmaxf(mx, __shfl_xor(mx, 16, 32));
    float s = 0.0f;
#pragma unroll
    for (int r = 0; r < 8; ++r) {
      const float e = __expf(at[t][r] - mx);
      at[t][r] = e;
      s += e;
    }
    s += __shfl_xor(s, 16, 32);
    const float inv = 1.0f / s;
#pragma unroll
    for (int r = 0; r < 8; ++r) at[t][r] *= inv;
  }

  // --- agg = sum_j attn * (vf + pos_enc)
#pragma unroll
  for (int t = 0; t < 8; ++t) {
    const int c = t * 16 + ln;
    float acc = 0.0f;
#pragma unroll
    for (int r = 0; r < 8; ++r)
      acc = fmaf(at[t][r], vfeat[rb[r] + c] + pos[t][r], acc);
    acc += __shfl_xor(acc, 16, 32);
    if (half == 0) agg[(size_t)p * DMODEL + c] = acc;
  }
}

// ---------------------------------------------------------------------------
// Launch: knn -> x=fc1 -> q/k/v -> fused attention -> fc2.
// Workspace layout (floats): x | q | k | v | agg | knn(int)
// ---------------------------------------------------------------------------
extern "C" void kernel_launch(void* const* d_in, const int* in_sizes, int n_in,
                              void* d_out, int out_size, void* d_ws, size_t ws_size,
                              hipStream_t stream) {
  const float* xyz   = (const float*)d_in[0];
  const float* feats = (const float*)d_in[1];
  const float* fc1_w = (const float*)d_in[2];
  const float* fc1_b = (const float*)d_in[3];
  const float* fc2_w = (const float*)d_in[4];
  const float* fc2_b = (const float*)d_in[5];
  const float* d1_w  = (const float*)d_in[6];
  const float* d1_b  = (const float*)d_in[7];
  const float* d2_w  = (const float*)d_in[8];
  const float* d2_b  = (const float*)d_in[9];
  const float* g1_w  = (const float*)d_in[10];
  const float* g1_b  = (const float*)d_in[11];
  const float* g2_w  = (const float*)d_in[12];
  const float* g2_b  = (const float*)d_in[13];
  const float* wq_w  = (const float*)d_in[14];
  const float* wk_w  = (const float*)d_in[15];
  const float* wv_w  = (const float*)d_in[16];

  const size_t SZ = ROWS * DMODEL;            // 4,194,304 floats
  float* ws  = (float*)d_ws;
  float* x   = ws;
  float* q   = x + SZ;
  float* k   = q + SZ;
  float* v   = k + SZ;
  float* agg = v + SZ;
  int*   knn = (int*)(agg + SZ);

  // KNN (independent of GEMMs, runs first)
  knn_kernel<<<dim3(NPTS / 256, BATCH, 1), 256, 0, stream>>>(xyz, knn);

  // x = features @ fc1 + b   (M=32768, K=64)
  wmma_gemm_rm<<<ROWS / 16 / 8, 256, 0, stream>>>(feats, fc1_w, fc1_b, x,
                                                  (int)ROWS, 64);
  // q, k, v = x @ W          (M=32768, K=128)
  wmma_gemm_rm<<<ROWS / 16 / 8, 256, 0, stream>>>(x, wq_w, nullptr, q, (int)ROWS, DMODEL);
  wmma_gemm_rm<<<ROWS / 16 / 8, 256, 0, stream>>>(x, wk_w, nullptr, k, (int)ROWS, DMODEL);
  wmma_gemm_rm<<<ROWS / 16 / 8, 256, 0, stream>>>(x, wv_w, nullptr, v, (int)ROWS, DMODEL);

  // Fused pos_enc + attention + aggregation (one wave per point)
  pt_attn_kernel<<<ROWS / 4, 128, 0, stream>>>(xyz, knn, q, k, v,
                                               d1_w, d1_b, d2_w, d2_b,
                                               g1_w, g1_b, g2_w, g2_b, agg);

  // out = agg @ fc2 + b
  wmma_gemm_rm<<<ROWS / 16 / 8, 256, 0, stream>>>(agg, fc2_w, fc2_b,
                                                  (float*)d_out, (int)ROWS, DMODEL);
}